// RandomElasticDeformation3D_22814866276555
// MI455X (gfx1250) — compile-verified
//
#include <hip/hip_runtime.h>
#include <math.h>

#define ALPHA 35.0f
#define NVOX (2 * 128 * 128 * 128)
#define IMGN (NVOX * 4)

typedef __attribute__((ext_vector_type(2))) float v2f;
typedef __attribute__((ext_vector_type(8))) float v8f;

// ---------------------------------------------------------------------------
// Kernel 1: build the combined 128x4 operator M = Smooth(16-tap gauss, zero
// pad (7,8)) o CubicResize(4 -> 128, Keys a=-0.5, half-pixel, normalized).
// M is identical for the d, h and w axes. Output: d_ws[0 .. 512) floats.
// ---------------------------------------------------------------------------
__global__ __launch_bounds__(128) void build_M_kernel(float* __restrict__ M) {
  __shared__ float R[128][4];
  const int o = threadIdx.x;  // 128 threads, one output row each

  // Keys cubic (a = -0.5) resize weights, JAX half-pixel convention
  const float s = (o + 0.5f) * (1.0f / 32.0f) - 0.5f;
  float w[4];
  float wsum = 0.0f;
#pragma unroll
  for (int i = 0; i < 4; ++i) {
    float t = fabsf(s - (float)i);
    float v;
    if (t <= 1.0f)
      v = (1.5f * t - 2.5f) * t * t + 1.0f;
    else if (t < 2.0f)
      v = -0.5f * (((t - 5.0f) * t + 8.0f) * t - 4.0f);
    else
      v = 0.0f;
    w[i] = v;
    wsum += v;
  }
  const float winv = 1.0f / wsum;  // JAX normalizes weights per output
#pragma unroll
  for (int i = 0; i < 4; ++i) R[o][i] = w[i] * winv;
  __syncthreads();

  // 16-tap gaussian: ax = m - 7, m = 0..15, sigma = 2.5
  float g[16];
  float gsum = 0.0f;
#pragma unroll
  for (int m = 0; m < 16; ++m) {
    float ax = (float)(m - 7);
    g[m] = __expf(-ax * ax * (1.0f / 12.5f));
    gsum += g[m];
  }
  const float ginv = 1.0f / gsum;

  // M[o][i] = sum_m g[m] * R[o + m - 7][i]  (zero outside [0,128))
#pragma unroll
  for (int i = 0; i < 4; ++i) {
    float acc = 0.0f;
#pragma unroll
    for (int m = 0; m < 16; ++m) {
      int p = o + m - 7;
      if (p >= 0 && p < 128) acc += g[m] * R[p][i];
    }
    M[o * 4 + i] = acc * ginv;
  }
}

// ---------------------------------------------------------------------------
// Kernel 2: fused flow reconstruction (WMMA) + trilinear image warp +
// nearest label warp. One block = one 16x16 (h,w) tile at fixed (b,d).
// ---------------------------------------------------------------------------
__device__ __forceinline__ float4 lerp4(float4 a, float4 b, float t) {
  float4 r;
  r.x = a.x * (1.0f - t) + b.x * t;
  r.y = a.y * (1.0f - t) + b.y * t;
  r.z = a.z * (1.0f - t) + b.z * t;
  r.w = a.w * (1.0f - t) + b.w * t;
  return r;
}

__global__ __launch_bounds__(256) void deform_kernel(
    const float4* __restrict__ img,   // (2,128,128,128,4) as float4 voxels
    const float* __restrict__ coarse, // (2,4,4,4,3)
    const int* __restrict__ label,    // (2,128,128,128)
    const float* __restrict__ M,      // 128x4
    float* __restrict__ out) {        // [IMGN image floats][NVOX label floats]
  __shared__ float lMh[16][4];
  __shared__ float lMw[16][4];
  __shared__ float lMd[4];
  __shared__ float sJK[3][4][4];       // s_c[j][k] = sum_i Md[i]*coarse[b,i,j,k,c]
  __shared__ float flowT[3][16][16];   // alpha-scaled flow tile per channel

  const int wt = blockIdx.x;           // w tile (0..7)
  const int ht = blockIdx.y;           // h tile (0..7)
  const int bz = blockIdx.z;           // b*128 + d
  const int b = bz >> 7;
  const int d = bz & 127;
  const int h0 = ht * 16;
  const int w0 = wt * 16;
  const int t = threadIdx.x;

  // Stage the rows of M this tile needs
  if (t < 64) {
    lMh[t >> 2][t & 3] = M[(h0 + (t >> 2)) * 4 + (t & 3)];
  } else if (t < 128) {
    int u = t - 64;
    lMw[u >> 2][u & 3] = M[(w0 + (u >> 2)) * 4 + (u & 3)];
  } else if (t < 132) {
    lMd[t - 128] = M[d * 4 + (t - 128)];
  }
  __syncthreads();

  // Contract the d axis: s_c = sum_i Md[i] * coarse[b,i,:,:,c]   (3x4x4)
  if (t < 48) {
    const int c = t >> 4;
    const int j = (t >> 2) & 3;
    const int k = t & 3;
    float acc = 0.0f;
#pragma unroll
    for (int i = 0; i < 4; ++i)
      acc += lMd[i] * coarse[(((b * 4 + i) * 4 + j) * 4 + k) * 3 + c];
    sJK[c][j][k] = acc;
  }
  __syncthreads();

  // Wave 0: per channel, F(16x16) = Mh(16x4) @ [ s_c(4x4) @ Mw(16x4)^T ](4x16)
  // Outer product via V_WMMA_F32_16X16X4_F32 (exact f32).
  if (t < 32) {
    const int half = t >> 4;   // lane group
    const int r = t & 15;
    const int k0 = half * 2;
    // A (16x4): lanes 0-15 hold K=0,1; lanes 16-31 hold K=2,3; M = lane&15
    v2f a;
    a.x = lMh[r][k0];
    a.y = lMh[r][k0 + 1];
#pragma unroll
    for (int c = 0; c < 3; ++c) {
      // B (4x16): lanes 0-15 hold rows K=0,1; lanes 16-31 rows K=2,3; N = lane&15
      float b0 = 0.0f, b1 = 0.0f;
#pragma unroll
      for (int kp = 0; kp < 4; ++kp) {
        b0 += sJK[c][k0][kp] * lMw[r][kp];
        b1 += sJK[c][k0 + 1][kp] * lMw[r][kp];
      }
      v2f bb;
      bb.x = b0;
      bb.y = b1;
      v8f acc = {};
      acc = __builtin_amdgcn_wmma_f32_16x16x4_f32(
          false, a, false, bb, (short)0, acc, false, false);
      // D layout: VGPR rr -> row rr + 8*half, col = lane&15
#pragma unroll
      for (int rr = 0; rr < 8; ++rr)
        flowT[c][rr + 8 * half][r] = ALPHA * acc[rr];
    }
  }
  __syncthreads();

  // Per-voxel warp
  const int hl = t >> 4;
  const int wl = t & 15;
  const int h = h0 + hl;
  const int w = w0 + wl;

  const float wd = (float)d + flowT[0][hl][wl];
  const float wh = (float)h + flowT[1][hl][wl];
  const float ww = (float)w + flowT[2][hl][wl];

  const float fd = floorf(wd), fh = floorf(wh), fw = floorf(ww);
  const float td = wd - fd, th = wh - fh, tw = ww - fw;

  const int d0 = min(max((int)fd, 0), 127);
  const int d1 = min(max((int)fd + 1, 0), 127);
  const int hh0 = min(max((int)fh, 0), 127);
  const int hh1 = min(max((int)fh + 1, 0), 127);
  const int ww0 = min(max((int)fw, 0), 127);
  const int ww1 = min(max((int)fw + 1, 0), 127);

  const size_t base = (size_t)b * (128 * 128 * 128);
#define GAT(di, hi, wi) img[base + (((size_t)(di) * 128 + (hi)) * 128 + (wi))]
  float4 c000 = GAT(d0, hh0, ww0);
  float4 c001 = GAT(d0, hh0, ww1);
  float4 c010 = GAT(d0, hh1, ww0);
  float4 c011 = GAT(d0, hh1, ww1);
  float4 c100 = GAT(d1, hh0, ww0);
  float4 c101 = GAT(d1, hh0, ww1);
  float4 c110 = GAT(d1, hh1, ww0);
  float4 c111 = GAT(d1, hh1, ww1);
#undef GAT

  float4 c00 = lerp4(c000, c001, tw);
  float4 c01 = lerp4(c010, c011, tw);
  float4 c10 = lerp4(c100, c101, tw);
  float4 c11 = lerp4(c110, c111, tw);
  float4 c0v = lerp4(c00, c01, th);
  float4 c1v = lerp4(c10, c11, th);
  float4 res = lerp4(c0v, c1v, td);

  const size_t vox = base + (((size_t)d * 128 + h) * 128 + w);
  ((float4*)out)[vox] = res;

  // Nearest-neighbor label warp (jnp.round == round-half-even == rintf)
  const int nd = min(max((int)rintf(wd), 0), 127);
  const int nh = min(max((int)rintf(wh), 0), 127);
  const int nw = min(max((int)rintf(ww), 0), 127);
  const int lab = label[base + (((size_t)nd * 128 + nh) * 128 + nw)];
  out[IMGN + vox] = (float)lab;
}

// ---------------------------------------------------------------------------
extern "C" void kernel_launch(void* const* d_in, const int* in_sizes, int n_in,
                              void* d_out, int out_size, void* d_ws,
                              size_t ws_size, hipStream_t stream) {
  (void)in_sizes;
  (void)n_in;
  (void)out_size;
  (void)ws_size;
  const float* img = (const float*)d_in[0];
  const float* coarse = (const float*)d_in[1];
  const int* label = (const int*)d_in[2];
  float* M = (float*)d_ws;  // 128*4 floats = 2 KB scratch

  build_M_kernel<<<1, 128, 0, stream>>>(M);

  dim3 grid(8, 8, 256);  // (w tiles, h tiles, b*128 + d)
  deform_kernel<<<grid, 256, 0, stream>>>((const float4*)img, coarse, label, M,
                                          (float*)d_out);
}